// SimpleGCN_4784593568413
// MI455X (gfx1250) — compile-verified
//
#include <hip/hip_runtime.h>
#include <hip/hip_bf16.h>
#include <math.h>

typedef __attribute__((ext_vector_type(2))) float v2f;
typedef __attribute__((ext_vector_type(8))) float v8f;

#define FDIM 128
#define LDST 132   // padded LDS stride for transposed W (bank-conflict-free b64 reads)

// ---------------------------------------------------------------------------
// H = X(nrows x 128) @ W(128 x 128), fp32 WMMA 16x16x4.
// One wave -> 16-row strip x all 128 cols (8 accumulator tiles).
// ---------------------------------------------------------------------------
__global__ __launch_bounds__(256) void k_gemm(const float* __restrict__ X,
                                              const float* __restrict__ W,
                                              float* __restrict__ Hout,
                                              int nrows)
{
    __shared__ float Wt[FDIM * LDST];
    // Stage W transposed: Wt[n*LDST + k] = W[k*128 + n]
    for (int idx = threadIdx.x; idx < FDIM * FDIM; idx += 256) {
        int k = idx >> 7, n = idx & 127;
        Wt[n * LDST + k] = W[idx];
    }
    __syncthreads();

    const int wave = threadIdx.x >> 5;
    const int lane = threadIdx.x & 31;
    const int row0 = (blockIdx.x * 8 + wave) * 16;
    if (row0 >= nrows) return;                       // wave-uniform: EXEC stays all-1s

    const int m     = lane & 15;                     // row within tile (A) / col (B,D)
    const int khalf = (lane >> 4) << 1;              // lanes 16-31 hold K=2,3
    const float* xrow = X + (size_t)(row0 + m) * FDIM + khalf;

    v8f acc[8] = {};                                 // 8 tiles of 16x16 f32

    for (int kk = 0; kk < 32; ++kk) {
        const v2f a = *(const v2f*)(xrow + kk * 4);  // A frag: X[row][4kk+khalf .. +1]
        #pragma unroll
        for (int t = 0; t < 8; ++t) {
            // B frag: W[4kk+khalf][n], W[4kk+khalf+1][n]  (contiguous in Wt)
            const v2f b = *(const v2f*)(&Wt[(t * 16 + m) * LDST + kk * 4 + khalf]);
            acc[t] = __builtin_amdgcn_wmma_f32_16x16x4_f32(
                false, a, false, b, (short)0, acc[t], false, false);
        }
    }

    // D layout: VGPR r -> M = r + 8*(lane>=16), N = (lane&15) + 16*t
    const int hi = lane >> 4;
    #pragma unroll
    for (int t = 0; t < 8; ++t) {
        #pragma unroll
        for (int r = 0; r < 8; ++r) {
            Hout[(size_t)(row0 + r + 8 * hi) * FDIM + t * 16 + m] = acc[t][r];
        }
    }
}

// ---------------------------------------------------------------------------
// Elementwise / graph kernels
// ---------------------------------------------------------------------------
__global__ void k_fill(float* __restrict__ p, float v, int n)
{
    int i = blockIdx.x * blockDim.x + threadIdx.x;
    if (i < n) p[i] = v;
}

__global__ void k_deg_edges(const int* __restrict__ dst,
                            const float* __restrict__ ew,
                            float* __restrict__ deg, int E)
{
    int i = blockIdx.x * blockDim.x + threadIdx.x;
    if (i < E) unsafeAtomicAdd(&deg[dst[i]], ew[i]);
}

__global__ void k_dinv(const float* __restrict__ deg, float* __restrict__ dinv, int n)
{
    int i = blockIdx.x * blockDim.x + threadIdx.x;
    if (i >= n) return;
    float d = deg[i];
    dinv[i] = (d > 0.0f) ? rsqrtf(fmaxf(d, 1e-12f)) : 0.0f;
}

// agg[i][:] = dinv[i]^2 * h[i][:]   (self-loop contribution, also re-inits agg)
__global__ void k_self(const float* __restrict__ Hm, const float* __restrict__ dinv,
                       float* __restrict__ agg, int n)
{
    int i = blockIdx.x * blockDim.x + threadIdx.x;
    int node = i >> 5, q = i & 31;
    if (node >= n) return;
    float s = dinv[node]; s *= s;
    float4 h = ((const float4*)(Hm + (size_t)node * FDIM))[q];
    float4 o = make_float4(s * h.x, s * h.y, s * h.z, s * h.w);
    ((float4*)(agg + (size_t)node * FDIM))[q] = o;
}

// One wave per edge: agg[dst] += dinv[src]*w*dinv[dst] * h[src]
__global__ __launch_bounds__(256) void k_scatter(const int* __restrict__ src,
                                                 const int* __restrict__ dst,
                                                 const float* __restrict__ ew,
                                                 const float* __restrict__ dinv,
                                                 const float* __restrict__ Hm,
                                                 float* __restrict__ agg, int E)
{
    const int lane = threadIdx.x & 31;
    int w = (blockIdx.x * blockDim.x + threadIdx.x) >> 5;
    const int nw = (gridDim.x * blockDim.x) >> 5;
    for (int e = w; e < E; e += nw) {
        const int s = src[e];
        const int d = dst[e];
        const float nrm = dinv[s] * ew[e] * dinv[d];
        const float4 h = ((const float4*)(Hm + (size_t)s * FDIM))[lane];
        float* out = agg + (size_t)d * FDIM + lane * 4;
        unsafeAtomicAdd(out + 0, nrm * h.x);
        unsafeAtomicAdd(out + 1, nrm * h.y);
        unsafeAtomicAdd(out + 2, nrm * h.z);
        unsafeAtomicAdd(out + 3, nrm * h.w);
        const int e2 = e + nw;                       // prefetch next gather row
        if (e2 < E) __builtin_prefetch(Hm + (size_t)src[e2] * FDIM + lane * 4, 0, 1);
    }
}

// x_out[i][:] = elu(agg[i][:] + b[:])
__global__ void k_bias_elu(const float* __restrict__ agg, const float* __restrict__ b,
                           float* __restrict__ xo, int n)
{
    int i = blockIdx.x * blockDim.x + threadIdx.x;
    int node = i >> 5, q = i & 31;
    if (node >= n) return;
    float4 a = ((const float4*)(agg + (size_t)node * FDIM))[q];
    const float4 bb = ((const float4*)b)[q];
    float v0 = a.x + bb.x, v1 = a.y + bb.y, v2 = a.z + bb.z, v3 = a.w + bb.w;
    float4 o;
    o.x = (v0 > 0.0f) ? v0 : expm1f(v0);
    o.y = (v1 > 0.0f) ? v1 : expm1f(v1);
    o.z = (v2 > 0.0f) ? v2 : expm1f(v2);
    o.w = (v3 > 0.0f) ? v3 : expm1f(v3);
    ((float4*)(xo + (size_t)node * FDIM))[q] = o;
}

__global__ void k_pool(const float* __restrict__ x, const int* __restrict__ batch,
                       float* __restrict__ sums, float* __restrict__ cnt, int n)
{
    int i = blockIdx.x * blockDim.x + threadIdx.x;
    int node = i >> 5, q = i & 31;
    if (node >= n) return;
    int g = batch[node];
    float4 v = ((const float4*)(x + (size_t)node * FDIM))[q];
    float* o = sums + (size_t)g * FDIM + q * 4;
    unsafeAtomicAdd(o + 0, v.x);
    unsafeAtomicAdd(o + 1, v.y);
    unsafeAtomicAdd(o + 2, v.z);
    unsafeAtomicAdd(o + 3, v.w);
    if (q == 0) unsafeAtomicAdd(cnt + g, 1.0f);
}

__global__ void k_head(const float* __restrict__ sums, const float* __restrict__ cnt,
                       const float* __restrict__ Wlin, const float* __restrict__ blin,
                       float* __restrict__ out, int G, int C)
{
    int i = blockIdx.x * blockDim.x + threadIdx.x;
    if (i >= G * C) return;
    int g = i / C, c = i % C;
    float inv = 1.0f / fmaxf(cnt[g], 1.0f);
    float acc = blin[c];
    #pragma unroll 4
    for (int k = 0; k < FDIM; ++k)
        acc += sums[(size_t)g * FDIM + k] * inv * Wlin[k * C + c];
    out[i] = acc;
}

// ---------------------------------------------------------------------------
extern "C" void kernel_launch(void* const* d_in, const int* in_sizes, int n_in,
                              void* d_out, int out_size, void* d_ws, size_t ws_size,
                              hipStream_t stream)
{
    const float* X     = (const float*)d_in[0];
    const int*   ei    = (const int*)d_in[1];
    const float* ea    = (const float*)d_in[2];
    const int*   batch = (const int*)d_in[4];
    const float* Wl[4] = {(const float*)d_in[5], (const float*)d_in[7],
                          (const float*)d_in[9], (const float*)d_in[11]};
    const float* bl[4] = {(const float*)d_in[6], (const float*)d_in[8],
                          (const float*)d_in[10], (const float*)d_in[12]};
    const float* Wlin  = (const float*)d_in[13];
    const float* blin  = (const float*)d_in[14];

    const int N = in_sizes[0] / FDIM;
    const int E = in_sizes[1] / 2;
    const int C = in_sizes[14];
    const int G = out_size / C;

    const int* srcI = ei;
    const int* dstI = ei + E;

    // Workspace layout
    char*  ws = (char*)d_ws;
    size_t NB = (size_t)N * FDIM * sizeof(float);
    auto a256 = [](size_t v) { return (v + 255) & ~(size_t)255; };
    size_t off = 0;
    float* bufA = (float*)(ws + off); off += a256(NB);
    float* bufB = (float*)(ws + off); off += a256(NB);
    float* bufC = (float*)(ws + off); off += a256(NB);
    float* deg  = (float*)(ws + off); off += a256((size_t)N * sizeof(float));
    float* dinv = (float*)(ws + off); off += a256((size_t)N * sizeof(float));
    float* sums = (float*)(ws + off); off += (size_t)G * FDIM * sizeof(float);
    float* cnt  = (float*)(ws + off); off += (size_t)G * sizeof(float);
    (void)ws_size;

    const int nodeBlocks = (N * 32 + 255) / 256;
    const int gemmBlocks = (N + 127) / 128;

    // Degree / normalization (shared by all 4 layers)
    k_fill<<<(N + 255) / 256, 256, 0, stream>>>(deg, 1.0f, N);         // self-loop weight
    k_deg_edges<<<(E + 255) / 256, 256, 0, stream>>>(dstI, ea, deg, E);
    k_dinv<<<(N + 255) / 256, 256, 0, stream>>>(deg, dinv, N);

    const float* xin = X;
    for (int L = 0; L < 4; ++L) {
        k_gemm<<<gemmBlocks, 256, 0, stream>>>(xin, Wl[L], bufB, N);
        k_self<<<nodeBlocks, 256, 0, stream>>>(bufB, dinv, bufA, N);
        k_scatter<<<4096, 256, 0, stream>>>(srcI, dstI, ea, dinv, bufB, bufA, E);
        k_bias_elu<<<nodeBlocks, 256, 0, stream>>>(bufA, bl[L], bufC, N);
        xin = bufC;
    }

    // Mean pool over graphs + linear head (sums and cnt are contiguous -> one fill)
    k_fill<<<(G * FDIM + G + 255) / 256, 256, 0, stream>>>(sums, 0.0f, G * FDIM + G);
    k_pool<<<nodeBlocks, 256, 0, stream>>>(xin, batch, sums, cnt, N);
    k_head<<<(G * C + 255) / 256, 256, 0, stream>>>(sums, cnt, Wlin, blin,
                                                    (float*)d_out, G, C);
}